// RankingModel_19816979104210
// MI455X (gfx1250) — compile-verified
//
#include <hip/hip_runtime.h>

#define BATCH 64
#define RROWS 4096
#define COLS  16
#define HID   32
#define NBLK  256
#define CAPACITY 16.0f
#define NEGV  -1e9f

typedef float v2f __attribute__((ext_vector_type(2)));
typedef float v8f __attribute__((ext_vector_type(8)));

// D = A(16x4 f32) * B(4x16 f32) + C(16x16 f32), wave32 WMMA.
__device__ __forceinline__ v8f wmma4(v2f a, v2f b, v8f c) {
  return __builtin_amdgcn_wmma_f32_16x16x4_f32(
      /*neg_a=*/false, a, /*neg_b=*/false, b,
      /*c_mod=*/(short)0, c, /*reuse_a=*/false, /*reuse_b=*/false);
}

__device__ __forceinline__ float gumbelize(float u) {
  return -__logf(-__logf(u));          // v_log_f32 path; argmax only needs relative accuracy
}

__global__ __launch_bounds__(256)
void ranking_fused_kernel(const float* __restrict__ table,   // [B,R,16]
                          const float* __restrict__ W1,      // [16,32]
                          const float* __restrict__ b1,      // [32]
                          const float* __restrict__ W2,      // [32,256]
                          const float* __restrict__ b2,      // [256]
                          const float* __restrict__ gum,     // [B,R,256] uniforms
                          float* __restrict__ out)           // [B,R,256] one-hots
{
  __shared__ float sW1[COLS * HID];                 // 2 KB
  __shared__ float sB1[HID];
  __shared__ float sW2[HID * NBLK];                 // 32 KB
  __shared__ float sB2[NBLK];
  __shared__ float sH[16 * HID];                    // hidden tile, 2 KB
  __shared__ __align__(16) float sL[16 * NBLK];     // logits(+gumbel) tile, 16 KB
  __shared__ float sVal[8];
  __shared__ int   sIdx[8];
  __shared__ int   sWin;

  const int tid  = threadIdx.x;
  const int lane = tid & 31;
  const int wave = tid >> 5;
  const int bidx = blockIdx.x;                      // batch sequence

  for (int i = tid; i < COLS * HID; i += 256) sW1[i] = W1[i];
  for (int i = tid; i < HID;        i += 256) sB1[i] = b1[i];
  for (int i = tid; i < HID * NBLK; i += 256) sW2[i] = W2[i];
  for (int i = tid; i < NBLK;       i += 256) sB2[i] = b2[i];
  __syncthreads();

  // WMMA fragment coordinates (wave32, 16x16x4 f32):
  //  A: lane&15 = M row; VGPR0/1 hold K={0,1} (lanes 0-15) or K={2,3} (lanes 16-31)
  //  B: VGPR0 holds rows K=0|2 striped across lane halves; VGPR1 rows K=1|3
  //  C/D: VGPR i, lane l -> row i + 8*(l>>4), col l&15
  const int mrow  = lane & 15;
  const int kHalf = (lane >> 4) << 1;               // 0 or 2
  const int ncol  = lane & 15;
  const int crow0 = (lane >> 4) << 3;               // 0 or 8

  float cnt = 0.0f;                                 // pick count for block id == tid
  const size_t seqBase = (size_t)bidx * RROWS;

  for (int t = 0; t < RROWS / 16; ++t) {
    const size_t row0 = seqBase + (size_t)t * 16;
    const float* gt = gum + row0 * NBLK;            // this tile's 16x256 gumbel uniforms

    // Issue the whole tile's gumbel loads early (b128); they complete under the WMMAs.
    float4 uu[4];
#pragma unroll
    for (int it = 0; it < 4; ++it)
      uu[it] = ((const float4*)gt)[tid + 256 * it];

    // Warm L2/WGP$ for the next tile (global_prefetch_b8); 256 thr * 64B = 16 KB.
    if (t + 1 < RROWS / 16)
      __builtin_prefetch(gt + 16 * NBLK + tid * 16, 0, 1);

    // ---- GEMM1 (wave 0): h = relu(table_tile[16x16] @ W1[16x32] + b1) -> sH
    if (wave == 0) {
      const float* At = table + (row0 + (size_t)mrow) * COLS;
      v8f acc0 = {0,0,0,0,0,0,0,0};
      v8f acc1 = {0,0,0,0,0,0,0,0};
      for (int c = 0; c < 4; ++c) {
        const int k0 = 4 * c;
        v2f a;
        a.x = At[k0 + kHalf + 0];
        a.y = At[k0 + kHalf + 1];
        v2f bm0, bm1;
        bm0.x = sW1[(k0 + kHalf + 0) * HID + 0  + ncol];
        bm0.y = sW1[(k0 + kHalf + 1) * HID + 0  + ncol];
        bm1.x = sW1[(k0 + kHalf + 0) * HID + 16 + ncol];
        bm1.y = sW1[(k0 + kHalf + 1) * HID + 16 + ncol];
        acc0 = wmma4(a, bm0, acc0);
        acc1 = wmma4(a, bm1, acc1);
      }
      const float bias0 = sB1[ncol];
      const float bias1 = sB1[16 + ncol];
      for (int i = 0; i < 8; ++i) {
        float v0 = acc0[i] + bias0; v0 = v0 > 0.f ? v0 : 0.f;
        float v1 = acc1[i] + bias1; v1 = v1 > 0.f ? v1 : 0.f;
        sH[(crow0 + i) * HID + 0  + ncol] = v0;
        sH[(crow0 + i) * HID + 16 + ncol] = v1;
      }
    }
    __syncthreads();

    // ---- GEMM2 (all waves): logits = relu(h[16x32] @ W2[32x256] + b2) -> sL
    v2f a2[8];
    for (int c = 0; c < 8; ++c) {
      const int k0 = 4 * c;
      a2[c].x = sH[mrow * HID + k0 + kHalf + 0];
      a2[c].y = sH[mrow * HID + k0 + kHalf + 1];
    }
    for (int nn = 0; nn < 2; ++nn) {
      const int n = wave * 2 + nn;                  // this wave's 16-wide N tile
      v8f acc = {0,0,0,0,0,0,0,0};
      for (int c = 0; c < 8; ++c) {
        const int k0 = 4 * c;
        v2f bm;
        bm.x = sW2[(k0 + kHalf + 0) * NBLK + n * 16 + ncol];
        bm.y = sW2[(k0 + kHalf + 1) * NBLK + n * 16 + ncol];
        acc = wmma4(a2[c], bm, acc);
      }
      const float bias = sB2[n * 16 + ncol];
      for (int i = 0; i < 8; ++i) {
        float v = acc[i] + bias;
        v = v > 0.f ? v : 0.f;
        sL[(crow0 + i) * NBLK + n * 16 + ncol] = v;
      }
    }
    __syncthreads();

    // ---- Fold Gumbel noise into the staged logits: sL := logits + g (b128 LDS RMW).
#pragma unroll
    for (int it = 0; it < 4; ++it) {
      const int idx = tid + 256 * it;
      float4 l = ((float4*)sL)[idx];
      l.x += gumbelize(uu[it].x);
      l.y += gumbelize(uu[it].y);
      l.z += gumbelize(uu[it].z);
      l.w += gumbelize(uu[it].w);
      ((float4*)sL)[idx] = l;
    }
    __syncthreads();

    // ---- Serial capacity-masked argmax over the 16 staged rows (LDS-only loop).
    // softmax is monotone => argmax(softmax(x+g)) == argmax(x+g); forward p2 == one_hot.
    for (int rr = 0; rr < 16; ++rr) {
      const int j = tid;                            // candidate block id 0..255
      float val = sL[rr * NBLK + j];
      if (cnt >= CAPACITY) val += NEGV;             // capacity mask

      // wave32 argmax (ties -> lower index, matching jnp.argmax)
      float bv = val; int bi = j;
      for (int off = 16; off > 0; off >>= 1) {
        float ov = __shfl_xor(bv, off, 32);
        int   oi = __shfl_xor(bi, off, 32);
        if (ov > bv || (ov == bv && oi < bi)) { bv = ov; bi = oi; }
      }
      if (lane == 0) { sVal[wave] = bv; sIdx[wave] = bi; }
      __syncthreads();
      if (tid == 0) {
        float m = sVal[0]; int mi = sIdx[0];
        for (int w = 1; w < 8; ++w)
          if (sVal[w] > m || (sVal[w] == m && sIdx[w] < mi)) { m = sVal[w]; mi = sIdx[w]; }
        sWin = mi;
      }
      __syncthreads();
      const int win = sWin;
      sL[rr * NBLK + j] = (j == win) ? 1.0f : 0.0f; // row rr is dead; reuse as one-hot
      if (j == win) cnt += 1.0f;
    }
    __syncthreads();                                // order one-hot writes vs bulk read-out

    // ---- Stream the one-hot tile to HBM as b128 stores.
    float4* ot = (float4*)(out + row0 * NBLK);
#pragma unroll
    for (int it = 0; it < 4; ++it)
      ot[tid + 256 * it] = ((float4*)sL)[tid + 256 * it];
    // Next tile's gemm barriers order sH/sL reuse against these reads.
  }
}

extern "C" void kernel_launch(void* const* d_in, const int* in_sizes, int n_in,
                              void* d_out, int out_size, void* d_ws, size_t ws_size,
                              hipStream_t stream) {
  (void)in_sizes; (void)n_in; (void)out_size; (void)d_ws; (void)ws_size;
  const float* table = (const float*)d_in[0];
  const float* W1    = (const float*)d_in[1];
  const float* b1    = (const float*)d_in[2];
  const float* W2    = (const float*)d_in[3];
  const float* b2    = (const float*)d_in[4];
  const float* gum   = (const float*)d_in[5];
  float* out = (float*)d_out;
  ranking_fused_kernel<<<BATCH, 256, 0, stream>>>(table, W1, b1, W2, b2, gum, out);
}